// LTSCell_8899172237578
// MI455X (gfx1250) — compile-verified
//
#include <hip/hip_runtime.h>
#include <stdint.h>

// ---------------------------------------------------------------------------
// LTC cell (liquid time-constant RNN) for MI455X / gfx1250.
//
// ~1.2e9 sigmoid pairs over 36 sequential sweeps, ~3 MB resident data ->
// purely VALU/transcendental bound (HBM roofline irrelevant).  No
// factorizable matmul exists (the sigmoid couples b,j,u), so WMMA does not
// apply; the CDNA5-specific path is the Tensor Data Mover: each workgroup
// DMAs the 256 KB packed parameter table into its 320 KB LDS once
// (tensor_load_to_lds + s_wait_tensorcnt, confirmed in disasm) and runs all
// 36 sweeps out of LDS.
//
// Inner loop uses sigmoid(t) = 0.5 + 0.5*tanh(t/2):
//   contribution = 0.5*W + 0.5*W*tanh(arg),  arg = 0.5*sigma*(v-mu)
// The 0.5*W constants sum over j to per-u constants hoisted out of the loop,
// leaving 4 VALU / 1 transcendental (v_tanh_f32) per (j,u) pair.
// ---------------------------------------------------------------------------

#define UNITS 128
#define INSZ 64
#define BATCH 2048
#define SDE_UNFOLDS 6
#define ODE_UNFOLDS 6
#define L2E 1.4426950408889634f

typedef unsigned int uint32;
typedef uint32 u32x4 __attribute__((ext_vector_type(4)));
typedef int    i32x4 __attribute__((ext_vector_type(4)));
typedef int    i32x8 __attribute__((ext_vector_type(8)));

// LDS layout of the main kernel (dynamic shared memory):
//   [0, 256KB)            packed recurrent params, float4 P[u*128 + j]
//   [256KB, 256KB+8.1KB)  v tile, float V[voff(j) + bi], 16 batch elems
#define P_BYTES (UNITS * UNITS * 16)
// j-half offset of +16 floats avoids the systematic 2-way bank conflict
// between lanes b and b+16 (rows 64 apart always share banks otherwise).
__device__ __forceinline__ int voff(int j) { return j * 16 + ((j >> 6) << 4); }
#define V_FLOATS (127 * 16 + 16 + 16)
#define SMEM_BYTES (P_BYTES + V_FLOATS * 4)

__device__ __forceinline__ float fast_tanh(float x) {
#if __has_builtin(__builtin_amdgcn_tanhf)
  return __builtin_amdgcn_tanhf(x);        // v_tanh_f32 (gfx1250 TRANS op)
#elif __has_builtin(__builtin_amdgcn_tanh_f32)
  return __builtin_amdgcn_tanh_f32(x);     // alternate builtin spelling
#else
#warning "CDNA5 probe: no amdgcn tanh builtin found; using exp2/rcp fallback in inner loop"
  // tanh(x) = 1 - 2/(1 + exp2(2*log2e*x))
  float e = __builtin_amdgcn_exp2f(2.0f * L2E * x);
  return fmaf(-2.0f, __builtin_amdgcn_rcpf(1.0f + e), 1.0f);
#endif
}

// ---------------------------------------------------------------------------
// Main LTC recurrence. 128 blocks x 1024 threads (32 wave32 = 8 waves/SIMD;
// 1 WG per WGP because of the 264 KB LDS allocation). Block owns 16 batch
// elements.  lane = { bi = lane&15 (batch), jh = lane>>4 (j-half) }.
// Wave w owns output units u0 = 4*w .. 4*w+3.
// ---------------------------------------------------------------------------
#define UPW 4  // units per wave

__global__ __launch_bounds__(1024, 1)
void ltc_main_kernel(const float* __restrict__ states,
                     const float4* __restrict__ prec,   // packed params (global)
                     const float* __restrict__ wnum_s,
                     const float* __restrict__ wden_s,
                     const float* __restrict__ sumn,
                     const float* __restrict__ sumd,
                     const float* __restrict__ vleak,
                     const float* __restrict__ gleak,
                     const float* __restrict__ cmt,
                     float* __restrict__ out) {
  extern __shared__ char smem[];
  float4* __restrict__ P = (float4*)smem;
  float* __restrict__ V = (float*)(smem + P_BYTES);

  const int tid = threadIdx.x;
  const int wave = tid >> 5;
  const int lane = tid & 31;
  const int bi = lane & 15;
  const int jh = lane >> 4;
  const int u0 = wave * UPW;
  const int b = blockIdx.x * 16 + bi;

  // ---- TDM: DMA the 256 KB packed parameter table into LDS (wave 0) ----
  if (tid == 0) {
    uint64_t ga = (uint64_t)(uintptr_t)prec;
    uint32 lds_base = __builtin_amdgcn_groupstaticsize();  // dyn-LDS offset
    // D# group 0: count=1, lds_addr, 57-bit global_addr, type=2 ("image")
    u32x4 g0 = {1u,
                lds_base,
                (uint32)(ga & 0xffffffffu),
                (uint32)((ga >> 32) & 0x1ffffffu) | 0x80000000u};
    // D# group 1: data_size=8B; 1-D tensor of 32768 elements = 256 KB,
    // tile_dim0 = 32768, tile_dim1/2 unused, strides = 32768.
    i32x8 g1 = {(int)0x00030000u,   // data_size=3 (8B), no multicast/pad
                (int)0x80000000u,   // tensor_dim0 = 32768  (bits 79:48)
                (int)0x00010000u,   // tensor_dim1 = 1      (bits 111:80)
                (int)0x80000000u,   // tile_dim0   = 32768  (bits 127:112)
                0,                  // tile_dim1/2 = 0
                (int)0x00008000u,   // tensor_dim0_stride = 32768
                (int)0x80000000u,   // tensor_dim1_stride = 32768
                0};
    i32x4 gz = {0, 0, 0, 0};
#if defined(__clang_major__) && (__clang_major__ >= 23)
    i32x8 gz8 = {0, 0, 0, 0, 0, 0, 0, 0};
    __builtin_amdgcn_tensor_load_to_lds(g0, g1, gz, gz, gz8, 0);
#else
    __builtin_amdgcn_tensor_load_to_lds(g0, g1, gz, gz, 0);
#endif
    __builtin_amdgcn_s_wait_tensorcnt((short)0);
  }
  __syncthreads();  // params visible to all waves

  // ---- per-lane constants (UPW owned units) ----
  // sn/sd fold in the sensory terms AND the hoisted 0.5*W column sums.
  float sc[UPW], sn[UPW], sd[UPW], cmv[UPW], CG[UPW], GV[UPW], vc[UPW];
#pragma unroll
  for (int k = 0; k < UPW; ++k) {
    int u = u0 + k;
    sc[k] = states[b * UNITS + u];
    sn[k] = wnum_s[b * UNITS + u] + sumn[u];
    sd[k] = wden_s[b * UNITS + u] + sumd[u];
    float g = gleak[u];
    cmv[k] = cmt[u];
    CG[k] = cmv[k] + g;
    GV[k] = g * vleak[u];
  }

  const int jbase = jh << 6;  // 0 or 64

  for (int sde = 0; sde < SDE_UNFOLDS; ++sde) {
#pragma unroll
    for (int k = 0; k < UPW; ++k) vc[k] = sc[k];
    if (lane < 16) {
#pragma unroll
      for (int k = 0; k < UPW; ++k) V[voff(u0 + k) + bi] = vc[k];
    }
    __syncthreads();

    for (int ode = 0; ode < ODE_UNFOLDS; ++ode) {
      float an[UPW], ad[UPW];
#pragma unroll
      for (int k = 0; k < UPW; ++k) { an[k] = 0.f; ad[k] = 0.f; }

      for (int jj = 0; jj < 64; ++jj) {
        int j = jbase + jj;
        float vj = V[voff(j) + bi];
#pragma unroll
        for (int k = 0; k < UPW; ++k) {
          float4 p = P[(u0 + k) * UNITS + j];    // ds_load_b128
          float th = fast_tanh(fmaf(p.x, vj, p.y));
          ad[k] = fmaf(p.z, th, ad[k]);
          an[k] = fmaf(p.w, th, an[k]);
        }
      }
      __syncthreads();  // all V reads done before rewrite

#pragma unroll
      for (int k = 0; k < UPW; ++k) {
        float num = an[k] + __shfl_xor(an[k], 16, 32) + sn[k];
        float den = ad[k] + __shfl_xor(ad[k], 16, 32) + sd[k];
        float vnew = (fmaf(cmv[k], vc[k], GV[k]) + num) *
                     __builtin_amdgcn_rcpf(CG[k] + den);
        vc[k] = vnew;
      }
      if (lane < 16) {
#pragma unroll
        for (int k = 0; k < UPW; ++k) V[voff(u0 + k) + bi] = vc[k];
      }
      __syncthreads();
    }

#pragma unroll
    for (int k = 0; k < UPW; ++k) sc[k] += vc[k];
    // next SDE round only rewrites rows this wave owns; the last ODE barrier
    // already fenced all readers, so no extra barrier is needed here.
  }

  if (lane < 16) {
#pragma unroll
    for (int k = 0; k < UPW; ++k) out[b * UNITS + u0 + k] = sc[k];
  }
}

// ---------------------------------------------------------------------------
// Pack recurrent params (tanh form), all source arrays indexed [j,u]:
//   P[u*128+j] = { 0.5*sigma, -0.5*sigma*mu, 0.5*W, 0.5*W*erev }
// ---------------------------------------------------------------------------
__global__ void pack_rec_kernel(const float* __restrict__ mu,
                                const float* __restrict__ sg,
                                const float* __restrict__ W,
                                const float* __restrict__ er,
                                float4* __restrict__ out) {
  int t = blockIdx.x * 256 + threadIdx.x;  // t = u*128 + j
  if (t >= UNITS * UNITS) return;
  int u = t >> 7, j = t & 127;
  int s = j * UNITS + u;
  float hs = 0.5f * sg[s];
  float hw = 0.5f * W[s];
  float4 p;
  p.x = hs;
  p.y = -hs * mu[s];
  p.z = hw;
  p.w = hw * er[s];
  out[t] = p;
}

// Per-u column sums of the constant halves: sumd[u] = sum_j 0.5*W[j,u],
// sumn[u] = sum_j 0.5*W[j,u]*erev[j,u].  Hoisted out of all 36 sweeps.
__global__ void colsum_kernel(const float4* __restrict__ prec,
                              float* __restrict__ sumn,
                              float* __restrict__ sumd) {
  int u = threadIdx.x;  // 128 threads, 1 block
  float sn = 0.f, sd = 0.f;
  for (int j = 0; j < UNITS; ++j) {
    float4 p = prec[u * UNITS + j];
    sd += p.z;
    sn += p.w;
  }
  sumn[u] = sn;
  sumd[u] = sd;
}

// ---------------------------------------------------------------------------
// Pack sensory params (exp2 form), folding the input affine map x=in*iw+ib:
//   PS[i*128+u] = { -ssig*log2e*iw, -ssig*log2e*ib + ssig*smu*log2e, sW, sW*serev }
// sigmoid = rcp(1 + exp2(PS.x*in + PS.y))
// ---------------------------------------------------------------------------
__global__ void pack_sens_kernel(const float* __restrict__ smu,
                                 const float* __restrict__ ssg,
                                 const float* __restrict__ sW,
                                 const float* __restrict__ ser,
                                 const float* __restrict__ iw,
                                 const float* __restrict__ ib,
                                 float4* __restrict__ out) {
  int t = blockIdx.x * 256 + threadIdx.x;  // t = i*128 + u
  if (t >= INSZ * UNITS) return;
  int i = t >> 7, u = t & 127;
  int s = i * UNITS + u;
  float a = -L2E * ssg[s];
  float w = sW[s];
  float4 p;
  p.x = a * iw[i];
  p.y = a * ib[i] + L2E * ssg[s] * smu[s];
  p.z = w;
  p.w = w * ser[s];
  out[t] = p;
}

// ---------------------------------------------------------------------------
// Sensory precompute (unfold-invariant): w_num_s/w_den_s [B,U].
// Block = 128 threads (one per u), handles 8 batch elements.
// ---------------------------------------------------------------------------
__global__ void sensory_kernel(const float* __restrict__ in,
                               const float4* __restrict__ ps,
                               float* __restrict__ wnum,
                               float* __restrict__ wden) {
  const int u = threadIdx.x;       // 0..127
  const int b0 = blockIdx.x * 8;   // 256 blocks
  float an[8], ad[8];
#pragma unroll
  for (int bb = 0; bb < 8; ++bb) { an[bb] = 0.f; ad[bb] = 0.f; }
  for (int i = 0; i < INSZ; ++i) {
    float4 p = ps[i * UNITS + u];  // coalesced across u
#pragma unroll
    for (int bb = 0; bb < 8; ++bb) {
      float x = in[(b0 + bb) * INSZ + i];  // uniform -> scalarized
      float e = __builtin_amdgcn_exp2f(fmaf(p.x, x, p.y));
      float r = __builtin_amdgcn_rcpf(1.0f + e);
      ad[bb] = fmaf(p.z, r, ad[bb]);
      an[bb] = fmaf(p.w, r, an[bb]);
    }
  }
#pragma unroll
  for (int bb = 0; bb < 8; ++bb) {
    wnum[(b0 + bb) * UNITS + u] = an[bb];
    wden[(b0 + bb) * UNITS + u] = ad[bb];
  }
}

// ---------------------------------------------------------------------------
extern "C" void kernel_launch(void* const* d_in, const int* in_sizes, int n_in,
                              void* d_out, int out_size, void* d_ws, size_t ws_size,
                              hipStream_t stream) {
  (void)in_sizes; (void)n_in; (void)out_size; (void)ws_size;
  const float* inputs = (const float*)d_in[0];
  const float* states = (const float*)d_in[1];
  const float* iw     = (const float*)d_in[2];
  const float* ib     = (const float*)d_in[3];
  const float* smu    = (const float*)d_in[4];
  const float* ssg    = (const float*)d_in[5];
  const float* sW     = (const float*)d_in[6];
  const float* ser    = (const float*)d_in[7];
  const float* mu     = (const float*)d_in[8];
  const float* sg     = (const float*)d_in[9];
  const float* W      = (const float*)d_in[10];
  const float* er     = (const float*)d_in[11];
  const float* vleak  = (const float*)d_in[12];
  const float* gleak  = (const float*)d_in[13];
  const float* cmt    = (const float*)d_in[14];
  float* out = (float*)d_out;

  // d_ws layout
  char* ws = (char*)d_ws;
  float4* prec  = (float4*)(ws);                          // 256 KB
  float4* psens = (float4*)(ws + P_BYTES);                // 128 KB
  float* wnum_s = (float*)(ws + P_BYTES + INSZ * UNITS * 16);
  float* wden_s = wnum_s + BATCH * UNITS;                 // 1 MB each
  float* sumn   = wden_s + BATCH * UNITS;                 // 512 B
  float* sumd   = sumn + UNITS;                           // 512 B

  // allow the 264 KB dynamic-LDS allocation (not a stream op; capture-safe)
  hipFuncSetAttribute((const void*)ltc_main_kernel,
                      hipFuncAttributeMaxDynamicSharedMemorySize, SMEM_BYTES);

  pack_rec_kernel<<<(UNITS * UNITS + 255) / 256, 256, 0, stream>>>(
      mu, sg, W, er, prec);
  colsum_kernel<<<1, UNITS, 0, stream>>>(prec, sumn, sumd);
  pack_sens_kernel<<<(INSZ * UNITS + 255) / 256, 256, 0, stream>>>(
      smu, ssg, sW, ser, iw, ib, psens);
  sensory_kernel<<<BATCH / 8, UNITS, 0, stream>>>(inputs, psens, wnum_s, wden_s);
  ltc_main_kernel<<<BATCH / 16, 1024, SMEM_BYTES, stream>>>(
      states, prec, wnum_s, wden_s, sumn, sumd, vleak, gleak, cmt, out);
}